// SelfAttention_74569222193169
// MI455X (gfx1250) — compile-verified
//
#include <hip/hip_runtime.h>
#include <math.h>

typedef __attribute__((ext_vector_type(2))) float v2f;
typedef __attribute__((ext_vector_type(8))) float v8f;

#define BATCH 2
#define SEQ   1024
#define DIM   1024
#define NH    16
#define HD    64
#define HALF  32
#define DEPTH 6

static __device__ __forceinline__ v8f wmma4(v2f a, v2f b, v8f c) {
  // V_WMMA_F32_16X16X4_F32 : D = A(16x4,f32) * B(4x16,f32) + C(16x16,f32)
  return __builtin_amdgcn_wmma_f32_16x16x4_f32(
      /*neg_a=*/false, a, /*neg_b=*/false, b,
      /*c_mod=*/(short)0, c, /*reuse_a=*/false, /*reuse_b=*/false);
}

// ---------------------------------------------------------------------------
// C[M x N] = A[M x K] @ B[K x N], row-major.
// One wave computes a 32x32 macro-tile = 2x2 WMMA tiles (4 independent
// accumulator chains). Inner K loop is software-pipelined: fragments for
// step k+4 are loaded into fresh registers before the WMMAs of step k, so
// the scheduler overlaps loads with matrix ops instead of draining loadcnt.
// ---------------------------------------------------------------------------
__global__ void gemm32_kernel(const float* __restrict__ A,
                              const float* __restrict__ B,
                              float* __restrict__ C,
                              int M, int N, int K) {
  const int lane = threadIdx.x & 31;
  const int wid  = threadIdx.x >> 5;
  const int tilesN = N >> 5;                     // 32-wide macro tiles
  const int w  = blockIdx.x * 8 + wid;           // wave-uniform
  const int tm = w / tilesN;
  const int tn = w % tilesN;
  if (tm >= (M >> 5)) return;                    // uniform per wave

  const int ln   = lane & 15;
  const int half = lane >> 4;
  const int kb   = half << 1;                    // K offset 0 or 2

  const float* a0 = A + (size_t)(tm * 32 + ln) * K + kb;   // rows 0..15
  const float* a1 = a0 + (size_t)16 * K;                   // rows 16..31
  const float* b0 = B + (size_t)kb * N + tn * 32 + ln;     // cols 0..15
  const float* b1 = b0 + 16;                               // cols 16..31

  v8f acc00 = {}, acc01 = {}, acc10 = {}, acc11 = {};

  // ---- pipeline prologue: fragments for k = 0 ----
  v2f af0 = *(const v2f*)a0;
  v2f af1 = *(const v2f*)a1;
  v2f bf0, bf1;
  bf0.x = b0[0]; bf0.y = b0[N];
  bf1.x = b1[0]; bf1.y = b1[N];

#pragma unroll 2
  for (int k = 0; k < K - 4; k += 4) {
    // issue next step's loads into fresh registers
    a0 += 4; a1 += 4;
    b0 += (size_t)4 * N; b1 += (size_t)4 * N;
    v2f naf0 = *(const v2f*)a0;
    v2f naf1 = *(const v2f*)a1;
    v2f nbf0, nbf1;
    nbf0.x = b0[0]; nbf0.y = b0[N];
    nbf1.x = b1[0]; nbf1.y = b1[N];

    // compute with current fragments (overlaps with the loads above)
    acc00 = wmma4(af0, bf0, acc00);
    acc01 = wmma4(af0, bf1, acc01);
    acc10 = wmma4(af1, bf0, acc10);
    acc11 = wmma4(af1, bf1, acc11);

    af0 = naf0; af1 = naf1; bf0 = nbf0; bf1 = nbf1;
  }
  // ---- pipeline epilogue ----
  acc00 = wmma4(af0, bf0, acc00);
  acc01 = wmma4(af0, bf1, acc01);
  acc10 = wmma4(af1, bf0, acc10);
  acc11 = wmma4(af1, bf1, acc11);

  const int colb = tn * 32 + ln;
  float* c0 = C + (size_t)(tm * 32 + half * 8) * N + colb;       // rows r / r+8
  float* c1 = c0 + (size_t)16 * N;                               // rows 16+r / 24+r
#pragma unroll
  for (int r = 0; r < 8; ++r) {
    c0[(size_t)r * N]      = acc00[r];
    c0[(size_t)r * N + 16] = acc01[r];
    c1[(size_t)r * N]      = acc10[r];
    c1[(size_t)r * N + 16] = acc11[r];
  }
}

// ---------------------------------------------------------------------------
// RoPE instruction stream: one thread per (b, s, head); 64-dim vector in regs.
// Applies to both Q and K buffers in place.
// ---------------------------------------------------------------------------
__global__ void rope_kernel(float* __restrict__ Q, float* __restrict__ Kb,
                            const int* __restrict__ Hm, const int* __restrict__ Bm,
                            const float* __restrict__ fc, const float* __restrict__ fs) {
  __shared__ float c1[HALF], s1[HALF], c2[HALF], s2[HALF];
  if (threadIdx.x < HALF) {
    c1[threadIdx.x] = fc[HALF + threadIdx.x];        // freqs row 1
    s1[threadIdx.x] = fs[HALF + threadIdx.x];
    c2[threadIdx.x] = fc[2 * HALF + threadIdx.x];    // freqs row 2
    s2[threadIdx.x] = fs[2 * HALF + threadIdx.x];
  }
  __syncthreads();

  const int tid = blockIdx.x * blockDim.x + threadIdx.x;   // 0 .. 32767
  const int h = tid & 15;
  const int s = (tid >> 4) & (SEQ - 1);
  const int b = tid >> 14;
  const int ms = b * SEQ + s;

  float v[HD];
  for (int which = 0; which < 2; ++which) {
    float* base = (which ? Kb : Q) + ((size_t)(b * SEQ + s) * DIM) + h * HD;
#pragma unroll
    for (int i = 0; i < 16; ++i)
      *(float4*)(v + 4 * i) = *(const float4*)(base + 4 * i);

    for (int d = 0; d < DEPTH; ++d) {
      if (Hm[d * (BATCH * SEQ) + ms]) {                 // interleaved pairs
#pragma unroll
        for (int j = 0; j < HALF; ++j) {
          float re = v[2 * j], im = v[2 * j + 1];
          v[2 * j]     = re * c1[j] - im * s1[j];
          v[2 * j + 1] = re * s1[j] + im * c1[j];
        }
      }
      for (int bb = 0; bb < 2; ++bb) {
        if (Bm[(d * 2 + bb) * (BATCH * SEQ) + ms]) {    // half-split pairs
          const float* cc = bb ? c2 : c1;
          const float* ss = bb ? s2 : s1;
#pragma unroll
          for (int j = 0; j < HALF; ++j) {
            float re = v[j], im = v[j + HALF];
            v[j]        = re * cc[j] - im * ss[j];
            v[j + HALF] = re * ss[j] + im * cc[j];
          }
        }
      }
    }
#pragma unroll
    for (int i = 0; i < 16; ++i)
      *(float4*)(base + 4 * i) = *(float4*)(v + 4 * i);
  }
}

// ---------------------------------------------------------------------------
// Flash-attention: one wave per (b, h, 16-query tile). Per key tile all
// fragment loads are batched into clauses before the WMMA bursts.
// ---------------------------------------------------------------------------
__global__ void attn_kernel(const float* __restrict__ Q, const float* __restrict__ K,
                            const float* __restrict__ V, const int* __restrict__ masks,
                            float* __restrict__ O) {
  __shared__ float pbuf[8][16][17];

  const int lane = threadIdx.x & 31;
  const int wv   = threadIdx.x >> 5;
  const int w  = blockIdx.x * 8 + wv;          // 0 .. 2047
  const int b  = w >> 10;                      // / (NH * 64)
  const int h  = (w >> 6) & 15;
  const int m0 = (w & 63) << 4;

  const int ln   = lane & 15;
  const int half = lane >> 4;
  const int kb   = half << 1;

  const float* Qh = Q + ((size_t)(b * SEQ + m0) * DIM) + h * HD;
  const float* Kh = K + ((size_t)b * SEQ * DIM) + h * HD;
  const float* Vh = V + ((size_t)b * SEQ * DIM) + h * HD;

  v2f qa[16];                                   // Q tile in A-layout, reused 64x
#pragma unroll
  for (int c = 0; c < 16; ++c)
    qa[c] = *(const v2f*)(Qh + (size_t)ln * DIM + c * 4 + kb);

  float Mx[8], L[8];
  v8f o0 = {}, o1 = {}, o2 = {}, o3 = {};
#pragma unroll
  for (int r = 0; r < 8; ++r) { Mx[r] = -INFINITY; L[r] = 0.f; }

  // pre-biased pointers bumped per key tile (immediate-offset friendly)
  const float* kcol  = Kh + (size_t)ln * DIM + kb;           // key = kt + ln
  const float* vrow  = Vh + (size_t)kb * DIM + ln;           // key row kt + kb
  const int*   mbase = masks + ((size_t)(b * SEQ) + m0 + 8 * half) * SEQ + ln;

  for (int kt = 0; kt < SEQ; kt += 16) {
    // ---- batch-load all K fragments for this tile (one load clause) ----
    v2f bf[16];
#pragma unroll
    for (int c = 0; c < 16; ++c)
      bf[c] = *(const v2f*)(kcol + c * 4);

    // ---- scores S = Q * K^T : two independent accumulator chains ----
    v8f sf0 = {}, sf1 = {};
#pragma unroll
    for (int c = 0; c < 8; ++c) {
      sf0 = wmma4(qa[2 * c],     bf[2 * c],     sf0);
      sf1 = wmma4(qa[2 * c + 1], bf[2 * c + 1], sf1);
    }

    // ---- mask + online softmax ----
    float alpha[8];
#pragma unroll
    for (int r = 0; r < 8; ++r) {
      float sv = (sf0[r] + sf1[r]) * 0.125f;    // 1/sqrt(64)
      if (mbase[(size_t)r * SEQ + kt] == 0) sv = -1.0e9f;
      float rm = sv;
      rm = fmaxf(rm, __shfl_xor(rm, 1));
      rm = fmaxf(rm, __shfl_xor(rm, 2));
      rm = fmaxf(rm, __shfl_xor(rm, 4));
      rm = fmaxf(rm, __shfl_xor(rm, 8));
      float mnew = fmaxf(Mx[r], rm);
      alpha[r] = __expf(Mx[r] - mnew);
      Mx[r] = mnew;
      float pv = __expf(sv - mnew);
      float rs = pv;
      rs += __shfl_xor(rs, 1);
      rs += __shfl_xor(rs, 2);
      rs += __shfl_xor(rs, 4);
      rs += __shfl_xor(rs, 8);
      L[r] = L[r] * alpha[r] + rs;
      pbuf[wv][r + 8 * half][ln] = pv;          // stage P for layout conversion
      o0[r] *= alpha[r]; o1[r] *= alpha[r]; o2[r] *= alpha[r]; o3[r] *= alpha[r];
    }
    __builtin_amdgcn_wave_barrier();            // keep DS store->load order

    // ---- batch-load P (LDS) and V fragments, then WMMA burst ----
    v2f pa[4];
#pragma unroll
    for (int c2 = 0; c2 < 4; ++c2) {
      pa[c2].x = pbuf[wv][ln][c2 * 4 + kb];
      pa[c2].y = pbuf[wv][ln][c2 * 4 + kb + 1];
    }
    v2f vb[4][4];
#pragma unroll
    for (int c2 = 0; c2 < 4; ++c2) {
      const float* vr0 = vrow + (size_t)(c2 * 4) * DIM;
      const float* vr1 = vr0 + DIM;
#pragma unroll
      for (int t = 0; t < 4; ++t) {
        vb[c2][t].x = vr0[t * 16];
        vb[c2][t].y = vr1[t * 16];
      }
    }
#pragma unroll
    for (int c2 = 0; c2 < 4; ++c2) {
      o0 = wmma4(pa[c2], vb[c2][0], o0);
      o1 = wmma4(pa[c2], vb[c2][1], o1);
      o2 = wmma4(pa[c2], vb[c2][2], o2);
      o3 = wmma4(pa[c2], vb[c2][3], o3);
    }
    __builtin_amdgcn_wave_barrier();

    kcol += (size_t)16 * DIM;
    vrow += (size_t)16 * DIM;
  }

  // ---- normalize and store (b, q, h, d) ----
#pragma unroll
  for (int r = 0; r < 8; ++r) {
    float inv = 1.0f / L[r];
    float* orow = O + ((size_t)(b * SEQ) + m0 + r + 8 * half) * DIM + h * HD;
    orow[0 * 16 + ln] = o0[r] * inv;
    orow[1 * 16 + ln] = o1[r] * inv;
    orow[2 * 16 + ln] = o2[r] * inv;
    orow[3 * 16 + ln] = o3[r] * inv;
  }
}

// ---------------------------------------------------------------------------
extern "C" void kernel_launch(void* const* d_in, const int* in_sizes, int n_in,
                              void* d_out, int out_size, void* d_ws, size_t ws_size,
                              hipStream_t stream) {
  (void)in_sizes; (void)n_in; (void)out_size; (void)ws_size;
  const float* x     = (const float*)d_in[0];
  const int*   masks = (const int*)d_in[1];
  const int*   Hm    = (const int*)d_in[2];
  const int*   Bm    = (const int*)d_in[3];
  const float* fc    = (const float*)d_in[4];
  const float* fs    = (const float*)d_in[5];
  const float* wq    = (const float*)d_in[6];
  const float* wk    = (const float*)d_in[7];
  const float* wvp   = (const float*)d_in[8];
  const float* wo    = (const float*)d_in[9];
  float* out = (float*)d_out;

  const size_t NE = (size_t)BATCH * SEQ * DIM;   // 2M floats each
  float* Qb = (float*)d_ws;
  float* Kb = Qb + NE;
  float* Vb = Kb + NE;
  float* Ab = Vb + NE;

  const int M = BATCH * SEQ, N = DIM, Kd = DIM;
  dim3 blk(256);
  const int gemmBlocks = (M / 32) * (N / 32) / 8;        // 256
  const int ropeBlocks = (BATCH * SEQ * NH) / 256;       // 128
  const int attnBlocks = (BATCH * NH * (SEQ / 16)) / 8;  // 256

  gemm32_kernel<<<gemmBlocks, blk, 0, stream>>>(x, wq, Qb, M, N, Kd);
  gemm32_kernel<<<gemmBlocks, blk, 0, stream>>>(x, wk, Kb, M, N, Kd);
  gemm32_kernel<<<gemmBlocks, blk, 0, stream>>>(x, wvp, Vb, M, N, Kd);
  rope_kernel<<<ropeBlocks, blk, 0, stream>>>(Qb, Kb, Hm, Bm, fc, fs);
  attn_kernel<<<attnBlocks, blk, 0, stream>>>(Qb, Kb, Vb, masks, Ab);
  gemm32_kernel<<<gemmBlocks, blk, 0, stream>>>(Ab, wo, out, M, N, Kd);
}